// MultiScaleSSIMLoss_6743098655729
// MI455X (gfx1250) — compile-verified
//
#include <hip/hip_runtime.h>

typedef float v2f __attribute__((ext_vector_type(2)));
typedef float v8f __attribute__((ext_vector_type(8)));
// 4-byte-aligned float pair for (possibly) misaligned b64 global loads
typedef float v2fu __attribute__((ext_vector_type(2), aligned(4)));

#define SSIM_C1 1.0e-4f
#define SSIM_C2 9.0e-4f

// 11-tap Gaussian, sigma=1.5, normalized (matches reference window)
__constant__ float G11[11] = {
    0.00102838f, 0.00759879f, 0.03600070f, 0.10936069f, 0.21300553f,
    0.26601172f, 0.21300553f, 0.10936069f, 0.03600070f, 0.00759879f,
    0.00102838f};

static __device__ __forceinline__ v2f mk2(float a, float b) {
  v2f r; r.x = a; r.y = b; return r;
}

// D(16x16 f32) = A(16x4 f32) * B(4x16 f32) + C
static __device__ __forceinline__ v8f wmma4(v2f a, v2f b, v8f c) {
  return __builtin_amdgcn_wmma_f32_16x16x4_f32(
      /*neg_a=*/false, a, /*neg_b=*/false, b,
      /*c_mod=*/(short)0, c, /*reuse_a=*/false, /*reuse_b=*/false);
}

// Branch-free zero padding: unconditional clamped load + select.
static __device__ __forceinline__ float loadClamp(const float* __restrict__ p,
                                                  int r, int c, int H, int W) {
  const int rcl = min(max(r, 0), H - 1);
  const int ccl = min(max(c, 0), W - 1);
  const float v = p[(ptrdiff_t)rcl * W + ccl];
  const bool ok = ((unsigned)r < (unsigned)H) && ((unsigned)c < (unsigned)W);
  return ok ? v : 0.0f;
}

// Stage 1: horizontal blur of 5 channels as G1 = band^T * I^T on WMMA.
// A = band (per-lane g[K - lane%16]), B = image slab fragment
// (N = lane%16 slab row, K = 2*half+v slab col), accumulate over 7 K-chunks.
template <bool INTERIOR>
static __device__ __forceinline__ void ssim_stage1(
    const float* __restrict__ Xb, const float* __restrict__ Yb, int H, int W,
    int r0, int c0, const float* __restrict__ band,
    float (*__restrict__ Hb)[32][20], int n16, int half) {
#pragma unroll
  for (int rc = 0; rc < 2; ++rc) {
    v8f A5[5] = {};
    const int row = r0 - 5 + n16 + 16 * rc;
    const float* __restrict__ xr = Xb + (ptrdiff_t)row * W;
    const float* __restrict__ yr = Yb + (ptrdiff_t)row * W;
#pragma unroll
    for (int q = 0; q < 7; ++q) {
      const int col = c0 - 5 + 4 * q + 2 * half;
      float x0, x1, y0, y1;
      if (INTERIOR) {
        const v2fu xv = *(const v2fu*)(xr + col);  // one b64 per pair
        const v2fu yv = *(const v2fu*)(yr + col);
        x0 = xv.x; x1 = xv.y;
        y0 = yv.x; y1 = yv.y;
      } else {
        x0 = loadClamp(Xb, row, col, H, W);
        x1 = loadClamp(Xb, row, col + 1, H, W);
        y0 = loadClamp(Yb, row, col, H, W);
        y1 = loadClamp(Yb, row, col + 1, H, W);
      }
      const v2f Ab = mk2(band[2 * q], band[2 * q + 1]);
      A5[0] = wmma4(Ab, mk2(x0, x1), A5[0]);            // blur(x)
      A5[1] = wmma4(Ab, mk2(y0, y1), A5[1]);            // blur(y)
      A5[2] = wmma4(Ab, mk2(x0 * x0, x1 * x1), A5[2]);  // blur(x*x)
      A5[3] = wmma4(Ab, mk2(y0 * y0, y1 * y1), A5[3]);  // blur(y*y)
      A5[4] = wmma4(Ab, mk2(x0 * y0, x1 * y1), A5[4]);  // blur(x*y)
    }
    // D element v lives at (n = v + 8*half, r = lane%16 + 16*rc)
    const int r = n16 + 16 * rc;
#pragma unroll
    for (int ch = 0; ch < 5; ++ch) {
      float* p = &Hb[ch][r][8 * half];
#pragma unroll
      for (int v = 0; v < 8; ++v) p[v] = A5[ch][v];
    }
  }
}

// One wave computes one 16x16 SSIM-map tile.
// Stage 2: O^T = G1 * band (vertical blur); A = G1 via LDS, B = band regs.
__global__ __launch_bounds__(128) void ssim_scale_kernel(
    const float* __restrict__ X, const float* __restrict__ Y, int H, int W,
    int txShift, int tpiShift, int totalTiles, float scaleFactor,
    double* __restrict__ accum) {
  __shared__ __align__(16) float Hbuf[4][5][32][20];  // [wave][ch][r][n]

  const int lane = threadIdx.x & 31;
  const int wv = threadIdx.y;
  const int n16 = lane & 15;
  const int half = lane >> 4;

  const int tile = blockIdx.x * 4 + wv;
  if (tile >= totalTiles) return;  // never taken (totals divisible by 4)

  const int img = tile >> tpiShift;
  const int rem = tile & ((1 << tpiShift) - 1);
  const int ty = rem >> txShift;
  const int tx = rem & ((1 << txShift) - 1);
  const int r0 = ty << 4, c0 = tx << 4;

  const float* __restrict__ Xb = X + (size_t)img * H * W;
  const float* __restrict__ Yb = Y + (size_t)img * H * W;

  // Band fragments: value at (K = 4q+2*half+v, lane%16) is g[K - lane%16].
  // Same per-lane values serve as stage-1 A operand and stage-2 B operand.
  float band[14];
#pragma unroll
  for (int q = 0; q < 7; ++q) {
#pragma unroll
    for (int v = 0; v < 2; ++v) {
      const int d = 4 * q + 2 * half + v - n16;
      const int di = min(max(d, 0), 10);
      const float g = G11[di];
      band[2 * q + v] = (d == di) ? g : 0.0f;
    }
  }

  // ---------------- Stage 1 (uniform interior/border split) ----------------
  const bool interior =
      (r0 >= 5) && (c0 >= 5) && (r0 + 27 <= H) && (c0 + 23 <= W);
  if (interior) {
    ssim_stage1<true>(Xb, Yb, H, W, r0, c0, band, Hbuf[wv], n16, half);
  } else {
    ssim_stage1<false>(Xb, Yb, H, W, r0, c0, band, Hbuf[wv], n16, half);
  }

  // ---------------- Stage 2: vertical blur ---------------------------------
  v8f O5[5] = {};
#pragma unroll
  for (int q = 0; q < 7; ++q) {
    const v2f Bb = mk2(band[2 * q], band[2 * q + 1]);
    const int k = 4 * q + 2 * half;  // A operand: M = lane%16, K = 2*half+v
#pragma unroll
    for (int ch = 0; ch < 5; ++ch) {
      const v2f Af = mk2(Hbuf[wv][ch][k][n16], Hbuf[wv][ch][k + 1][n16]);
      O5[ch] = wmma4(Af, Bb, O5[ch]);
    }
  }

  // ---------------- SSIM map + reduction -----------------------------------
  float s = 0.0f;
#pragma unroll
  for (int v = 0; v < 8; ++v) {
    const float mu1 = O5[0][v], mu2 = O5[1][v];
    const float bxx = O5[2][v], byy = O5[3][v], bxy = O5[4][v];
    const float m11 = mu1 * mu1, m22 = mu2 * mu2, m12 = mu1 * mu2;
    const float num = (2.0f * m12 + SSIM_C1) * (2.0f * (bxy - m12) + SSIM_C2);
    const float den =
        (m11 + m22 + SSIM_C1) * ((bxx - m11) + (byy - m22) + SSIM_C2);
    s += num / den;
  }
#pragma unroll
  for (int off = 16; off; off >>= 1) s += __shfl_xor(s, off, 32);
  if (lane == 0) atomicAdd(accum, (double)s * (double)scaleFactor);
}

// 2x2 average pool for both images at once (floor mode, stride 2)
__global__ void pool_kernel(const float* __restrict__ A,
                            const float* __restrict__ B, float* __restrict__ PA,
                            float* __restrict__ PB, int Wo, int hwShift,
                            int wShift, int total) {
  const int idx = blockIdx.x * blockDim.x + threadIdx.x;
  if (idx >= total) return;
  const int Wi = Wo * 2;
  const int n = idx >> hwShift;
  const int rem = idx & ((1 << hwShift) - 1);
  const int y = rem >> wShift;
  const int x = rem & (Wo - 1);
  const size_t base = (((size_t)n << (hwShift + 2)) /* n*Hi*Wi */) +
                      (size_t)(2 * y) * Wi + 2 * x;
  const v2fu a0 = *(const v2fu*)(A + base);
  const v2fu a1 = *(const v2fu*)(A + base + Wi);
  const v2fu b0 = *(const v2fu*)(B + base);
  const v2fu b1 = *(const v2fu*)(B + base + Wi);
  PA[idx] = 0.25f * ((a0.x + a0.y) + (a1.x + a1.y));
  PB[idx] = 0.25f * ((b0.x + b0.y) + (b1.x + b1.y));
}

__global__ void init_kernel(double* acc) { acc[0] = 0.0; }

__global__ void finalize_kernel(const double* __restrict__ acc,
                                float* __restrict__ out) {
  out[0] = (float)(1.0 - acc[0]);
}

extern "C" void kernel_launch(void* const* d_in, const int* in_sizes, int n_in,
                              void* d_out, int out_size, void* d_ws,
                              size_t ws_size, hipStream_t stream) {
  const float* pred = (const float*)d_in[0];
  const float* targ = (const float*)d_in[1];
  float* out = (float*)d_out;

  double* acc = (double*)d_ws;
  float* p1 = (float*)((char*)d_ws + 256);
  const int S1 = 64 * 256 * 256;  // 4,194,304
  const int S2 = 64 * 128 * 128;  // 1,048,576
  float* t1 = p1 + S1;
  float* p2 = t1 + S1;
  float* t2 = p2 + S2;

  init_kernel<<<1, 1, 0, stream>>>(acc);

  const dim3 blk(32, 4);
  {  // scale 0: 512x512, weight 0.5, tilesX=32 (shift 5), tiles/img=1024 (10)
    const int total = 64 * 1024;
    ssim_scale_kernel<<<dim3(total / 4), blk, 0, stream>>>(
        pred, targ, 512, 512, 5, 10, total, 0.5f / (64.0f * 512.0f * 512.0f),
        acc);
  }
  // pool 512 -> 256 : Hi*Wi = 2^18 per image, Ho*Wo = 2^16
  pool_kernel<<<dim3((S1 + 255) / 256), dim3(256), 0, stream>>>(
      pred, targ, p1, t1, 256, 16, 8, S1);
  {  // scale 1: 256x256, weight 0.3, tilesX=16 (shift 4), tiles/img=256 (8)
    const int total = 64 * 256;
    ssim_scale_kernel<<<dim3(total / 4), blk, 0, stream>>>(
        p1, t1, 256, 256, 4, 8, total, 0.3f / (64.0f * 256.0f * 256.0f), acc);
  }
  // pool 256 -> 128 : Ho*Wo = 2^14
  pool_kernel<<<dim3((S2 + 255) / 256), dim3(256), 0, stream>>>(
      p1, t1, p2, t2, 128, 14, 7, S2);
  {  // scale 2: 128x128, weight 0.2, tilesX=8 (shift 3), tiles/img=64 (6)
    const int total = 64 * 64;
    ssim_scale_kernel<<<dim3(total / 4), blk, 0, stream>>>(
        p2, t2, 128, 128, 3, 6, total, 0.2f / (64.0f * 128.0f * 128.0f), acc);
  }
  finalize_kernel<<<1, 1, 0, stream>>>(acc, out);
}